// RecurrentGCN_10728828305408
// MI455X (gfx1250) — compile-verified
//
#include <hip/hip_runtime.h>
#include <math.h>

// ---------------- CDNA5 WMMA types ----------------
typedef float v2f __attribute__((ext_vector_type(2)));
typedef float v8f __attribute__((ext_vector_type(8)));

#define N_NODES   8192
#define NTILES    (N_NODES / 16)   // 512
#define BATCH     8
#define CIN       4
#define CO        2
#define CH        6                // Ci + Co
#define COLS      48               // BATCH*CH
#define DDIM      4
#define WAVES_PER_NB 4             // waves cooperating on one 16-row n-block

// workspace layout (in floats)
#define WS_XB0   0
#define WS_XB1   (WS_XB0 + NTILES * 16 * 64)        // 524288
#define WS_P1    (WS_XB1 + NTILES * 16 * 64)        // 1048576
#define WS_P2    (WS_P1  + N_NODES * COLS)          // 1441792
#define WS_Z     (WS_P2  + N_NODES * COLS)          // 1835008
#define WS_HR    (WS_Z   + N_NODES * BATCH * CO)    // 1966080
// total = 2097152 floats = 8 MB

// XB layout: XB[(((mt*4 + kc)*4 + j)*32 + lane)*2 + p] =
//   Xg[m = mt*16 + kc*4 + (lane>>4)*2 + p][col = j*16 + (lane&15)]
// columns: col<48 -> (b=col/6, c=col%6) ; col==48 -> 1.0 (rowsum) ; col>48 -> 0

// ---------------------------------------------------------------------------
// Kernel 0: build XB0 (full) and XB1 (x part + ones column; h*R filled later)
// ---------------------------------------------------------------------------
__global__ void prep_kernel(const float* __restrict__ x,
                            const float* __restrict__ h,
                            float* __restrict__ xb0,
                            float* __restrict__ xb1) {
    int t = blockIdx.x * blockDim.x + threadIdx.x;
    if (t >= NTILES * 16 * 64) return;
    int p    =  t        & 1;
    int lane = (t >> 1)  & 31;
    int j    = (t >> 6)  & 3;
    int kc   = (t >> 8)  & 3;
    int mt   =  t >> 10;
    int m    = mt * 16 + kc * 4 + ((lane >> 4) * 2) + p;
    int col  = j * 16 + (lane & 15);

    float v0, v1;
    if (col < COLS) {
        int b = col / CH, c = col % CH;
        if (c < CIN) {
            float xv = x[((size_t)b * N_NODES + m) * CIN + c];
            v0 = xv; v1 = xv;                        // Xg1 shares the x part
        } else {
            v0 = h[((size_t)b * N_NODES + m) * CO + (c - CIN)];
            v1 = 0.0f;                               // gate kernel writes h*R
        }
    } else if (col == COLS) {
        v0 = 1.0f; v1 = 1.0f;                        // ones column -> rowsum
    } else {
        v0 = 0.0f; v1 = 0.0f;
    }
    xb0[t] = v0;
    xb1[t] = v1;
}

// ---------------------------------------------------------------------------
// Kernel 1/3: fused  P = softmax(relu(E E^T)) @ Xg
// 4 waves cooperate on one 16-row n-block; each takes every 4th m-tile and
// accumulates a partial 16x64 U tile; LDS tree-reduce + rowsum-normalize.
// ---------------------------------------------------------------------------
__global__ void __launch_bounds__(32 * WAVES_PER_NB)
agg_kernel(const float* __restrict__ e,
           const float* __restrict__ xb,
           float* __restrict__ P) {
    __shared__ float wtile[WAVES_PER_NB * 288];      // per-wave 16x18 W staging
    __shared__ float red[WAVES_PER_NB * 1024];       // per-wave 16x64 partials

    int lane = threadIdx.x & 31;
    int wave = threadIdx.x >> 5;
    int nb   = blockIdx.x;                           // n-block id (0..511)
    float* w = &wtile[wave * 288];

    int lh = lane >> 4;                              // half-wave: 0/1
    int ll = lane & 15;

    // A operand: E rows of this n-block; (M=ll, K = lh*2 + {0,1})
    v2f ea;
    {
        const float* p = e + ((size_t)(nb * 16 + ll)) * DDIM + lh * 2;
        ea[0] = p[0]; ea[1] = p[1];
    }

    v8f acc0 = {}, acc1 = {}, acc2 = {}, acc3 = {};

    // software-pipelined E-tile B operand: (K = lh*2 + {0,1}, N = ll)
    v2f eb;
    {
        const float* p = e + ((size_t)(wave * 16 + ll)) * DDIM + lh * 2;
        eb[0] = p[0]; eb[1] = p[1];
    }

    for (int it = 0; it < NTILES / WAVES_PER_NB; ++it) {
        int mt = it * WAVES_PER_NB + wave;

        // stage all 16 XB B-operands for this tile up front (overlaps the
        // score-WMMA + exp + LDS chain with global latency)
        const float* xbt = xb + (size_t)mt * 1024;   // 16x64 tile
        v2f b[16];
        #pragma unroll
        for (int q = 0; q < 16; ++q)
            b[q] = *(const v2f*)(xbt + ((size_t)(q * 32 + lane)) * 2);

        // prefetch next iteration's E tile (wrap mask keeps it in-bounds)
        int mtn = (mt + WAVES_PER_NB) & (NTILES - 1);
        v2f ebn;
        {
            const float* p = e + ((size_t)(mtn * 16 + ll)) * DDIM + lh * 2;
            ebn[0] = p[0]; ebn[1] = p[1];
        }

        // 16x16 score tile: S = E_n(16x4) @ E_m^T(4x16)
        v8f s = {};
        s = __builtin_amdgcn_wmma_f32_16x16x4_f32(false, ea, false, eb,
                                                  (short)0, s, false, false);
        // softmax numerator: exp(relu(s))  (scores are tiny; shift-free OK)
        #pragma unroll
        for (int r = 0; r < 8; ++r)
            s[r] = __expf(fmaxf(s[r], 0.0f));
        // D layout -> LDS row-major (stride 18: conflict-free, 8B aligned)
        #pragma unroll
        for (int r = 0; r < 8; ++r)
            w[(r + 8 * lh) * 18 + ll] = s[r];
        // compile-time ordering only; DS ops are in-order within a wave on
        // CDNA5, so no hardware wait is needed between store and load.
        asm volatile("" ::: "memory");

        #pragma unroll
        for (int kc = 0; kc < 4; ++kc) {
            // A operand of W: (M=ll, K = lh*2 + {0,1}) within chunk kc
            v2f aw = *(const v2f*)&w[ll * 18 + kc * 4 + lh * 2];
            acc0 = __builtin_amdgcn_wmma_f32_16x16x4_f32(false, aw, false,
                    b[kc * 4 + 0], (short)0, acc0, false, false);
            acc1 = __builtin_amdgcn_wmma_f32_16x16x4_f32(false, aw, false,
                    b[kc * 4 + 1], (short)0, acc1, false, false);
            acc2 = __builtin_amdgcn_wmma_f32_16x16x4_f32(false, aw, false,
                    b[kc * 4 + 2], (short)0, acc2, false, false);
            acc3 = __builtin_amdgcn_wmma_f32_16x16x4_f32(false, aw, false,
                    b[kc * 4 + 3], (short)0, acc3, false, false);
        }
        eb = ebn;
    }

    // dump partial 16x64 U tile (D layout: vgpr r -> row r+8*lh, col j*16+ll)
    #pragma unroll
    for (int r = 0; r < 8; ++r) {
        float* rp = &red[wave * 1024 + (r + 8 * lh) * 64 + ll];
        rp[0]  = acc0[r];
        rp[16] = acc1[r];
        rp[32] = acc2[r];
        rp[48] = acc3[r];
    }
    __syncthreads();

    // cross-wave reduce into wave0's region
    for (int k = threadIdx.x; k < 1024; k += 32 * WAVES_PER_NB)
        red[k] = red[k] + red[k + 1024] + red[k + 2048] + red[k + 3072];
    __syncthreads();

    // softmax denominator is the ones-column (col 48); normalize & store
    for (int k = threadIdx.x; k < 16 * COLS; k += 32 * WAVES_PER_NB) {
        int row = k / COLS, col = k - row * COLS;
        float v  = red[row * 64 + col];
        float rs = red[row * 64 + 48];
        P[((size_t)(nb * 16 + row)) * COLS + col] = v / rs;
    }
}

// ---------------------------------------------------------------------------
// Kernel 2: gate  ZR = sigmoid(avwgcn(...)) ; Z stored ; h*R scattered to XB1
// ---------------------------------------------------------------------------
__global__ void gate_kernel(const float* __restrict__ x,
                            const float* __restrict__ e,
                            const float* __restrict__ h,
                            const float* __restrict__ gw,   // (D,2,6,4)
                            const float* __restrict__ gb,   // (D,4)
                            const float* __restrict__ P1,
                            float* __restrict__ Z,
                            float* __restrict__ HR,
                            float* __restrict__ xb1) {
    int t = blockIdx.x * blockDim.x + threadIdx.x;
    if (t >= N_NODES * BATCH) return;
    int n = t & (N_NODES - 1);
    int b = t >> 13;

    float ev[DDIM];
    #pragma unroll
    for (int d = 0; d < DDIM; ++d) ev[d] = e[n * DDIM + d];

    float self[CH], agg[CH];
    #pragma unroll
    for (int i = 0; i < CIN; ++i) self[i] = x[((size_t)b * N_NODES + n) * CIN + i];
    self[4] = h[((size_t)b * N_NODES + n) * CO + 0];
    self[5] = h[((size_t)b * N_NODES + n) * CO + 1];
    #pragma unroll
    for (int i = 0; i < CH; ++i) agg[i] = P1[(size_t)n * COLS + b * CH + i];

    float zr[4];
    #pragma unroll
    for (int o = 0; o < 4; ++o) {
        float acc = 0.0f;
        #pragma unroll
        for (int d = 0; d < DDIM; ++d) acc += ev[d] * gb[d * 4 + o];
        #pragma unroll
        for (int i = 0; i < CH; ++i) {
            float w0 = 0.0f, w1 = 0.0f;
            #pragma unroll
            for (int d = 0; d < DDIM; ++d) {
                w0 += ev[d] * gw[((d * 2 + 0) * CH + i) * 4 + o];
                w1 += ev[d] * gw[((d * 2 + 1) * CH + i) * 4 + o];
            }
            acc += self[i] * w0 + agg[i] * w1;
        }
        zr[o] = 1.0f / (1.0f + __expf(-acc));
    }

    Z[((size_t)b * N_NODES + n) * CO + 0] = zr[0];
    Z[((size_t)b * N_NODES + n) * CO + 1] = zr[1];
    float hr0 = self[4] * zr[2];
    float hr1 = self[5] * zr[3];
    HR[((size_t)b * N_NODES + n) * CO + 0] = hr0;
    HR[((size_t)b * N_NODES + n) * CO + 1] = hr1;

    // scatter h*R into XB1 tile layout (cols b*6+4, b*6+5 of row n)
    int mt = n >> 4, loc = n & 15;
    int kc = loc >> 2, kl = loc & 3;
    int lh2 = kl >> 1, p = kl & 1;
    #pragma unroll
    for (int cc = 0; cc < 2; ++cc) {
        int col  = b * CH + CIN + cc;
        int j    = col >> 4;
        int lane = lh2 * 16 + (col & 15);
        xb1[((((size_t)mt * 4 + kc) * 4 + j) * 32 + lane) * 2 + p] = cc ? hr1 : hr0;
    }
}

// ---------------------------------------------------------------------------
// Kernel 4: update  C = tanh(avwgcn) ; h0 = Z*h + (1-Z)*C ; y = relu(h0)@Wl+bl
// ---------------------------------------------------------------------------
__global__ void update_kernel(const float* __restrict__ x,
                              const float* __restrict__ e,
                              const float* __restrict__ h,
                              const float* __restrict__ uw,   // (D,2,6,2)
                              const float* __restrict__ ub,   // (D,2)
                              const float* __restrict__ lw,   // (1,2)
                              const float* __restrict__ lb,   // (1,)
                              const float* __restrict__ P2,
                              const float* __restrict__ Z,
                              const float* __restrict__ HR,
                              float* __restrict__ out) {
    int t = blockIdx.x * blockDim.x + threadIdx.x;
    if (t >= N_NODES * BATCH) return;
    int n = t & (N_NODES - 1);
    int b = t >> 13;

    float ev[DDIM];
    #pragma unroll
    for (int d = 0; d < DDIM; ++d) ev[d] = e[n * DDIM + d];

    float self[CH], agg[CH];
    #pragma unroll
    for (int i = 0; i < CIN; ++i) self[i] = x[((size_t)b * N_NODES + n) * CIN + i];
    self[4] = HR[((size_t)b * N_NODES + n) * CO + 0];
    self[5] = HR[((size_t)b * N_NODES + n) * CO + 1];
    #pragma unroll
    for (int i = 0; i < CH; ++i) agg[i] = P2[(size_t)n * COLS + b * CH + i];

    float C[CO];
    #pragma unroll
    for (int o = 0; o < CO; ++o) {
        float acc = 0.0f;
        #pragma unroll
        for (int d = 0; d < DDIM; ++d) acc += ev[d] * ub[d * CO + o];
        #pragma unroll
        for (int i = 0; i < CH; ++i) {
            float w0 = 0.0f, w1 = 0.0f;
            #pragma unroll
            for (int d = 0; d < DDIM; ++d) {
                w0 += ev[d] * uw[((d * 2 + 0) * CH + i) * CO + o];
                w1 += ev[d] * uw[((d * 2 + 1) * CH + i) * CO + o];
            }
            acc += self[i] * w0 + agg[i] * w1;
        }
        C[o] = tanhf(acc);
    }

    float z0 = Z[((size_t)b * N_NODES + n) * CO + 0];
    float z1 = Z[((size_t)b * N_NODES + n) * CO + 1];
    float h0 = z0 * h[((size_t)b * N_NODES + n) * CO + 0] + (1.0f - z0) * C[0];
    float h1 = z1 * h[((size_t)b * N_NODES + n) * CO + 1] + (1.0f - z1) * C[1];

    float y = fmaxf(h0, 0.0f) * lw[0] + fmaxf(h1, 0.0f) * lw[1] + lb[0];

    out[(size_t)b * N_NODES + n] = y;                         // y (8,8192,1)
    float* oh = out + (size_t)BATCH * N_NODES;                // h0 (8,8192,2)
    oh[((size_t)b * N_NODES + n) * CO + 0] = h0;
    oh[((size_t)b * N_NODES + n) * CO + 1] = h1;
}

// ---------------------------------------------------------------------------
extern "C" void kernel_launch(void* const* d_in, const int* in_sizes, int n_in,
                              void* d_out, int out_size, void* d_ws, size_t ws_size,
                              hipStream_t stream) {
    (void)in_sizes; (void)n_in; (void)out_size; (void)ws_size;
    const float* x  = (const float*)d_in[0];
    const float* e  = (const float*)d_in[1];
    const float* h  = (const float*)d_in[2];
    const float* gw = (const float*)d_in[3];
    const float* gb = (const float*)d_in[4];
    const float* uw = (const float*)d_in[5];
    const float* ub = (const float*)d_in[6];
    const float* lw = (const float*)d_in[7];
    const float* lb = (const float*)d_in[8];
    float* out = (float*)d_out;

    float* ws  = (float*)d_ws;
    float* xb0 = ws + WS_XB0;
    float* xb1 = ws + WS_XB1;
    float* P1  = ws + WS_P1;
    float* P2  = ws + WS_P2;
    float* Zb  = ws + WS_Z;
    float* HR  = ws + WS_HR;

    // 0) build WMMA-B-operand-ready Xg buffers
    prep_kernel<<<(NTILES * 16 * 64) / 256, 256, 0, stream>>>(x, h, xb0, xb1);
    // 1) P1 = softmax(relu(E E^T)) @ [x,h]
    agg_kernel<<<NTILES, 32 * WAVES_PER_NB, 0, stream>>>(e, xb0, P1);
    // 2) gate: Z,R ; fill XB1 with h*R
    gate_kernel<<<(N_NODES * BATCH) / 256, 256, 0, stream>>>(x, e, h, gw, gb,
                                                             P1, Zb, HR, xb1);
    // 3) P2 = softmax(relu(E E^T)) @ [x, h*R]
    agg_kernel<<<NTILES, 32 * WAVES_PER_NB, 0, stream>>>(e, xb1, P2);
    // 4) update + GRU combine + linear head
    update_kernel<<<(N_NODES * BATCH) / 256, 256, 0, stream>>>(x, e, h, uw, ub,
                                                               lw, lb, P2, Zb,
                                                               HR, out);
}